// WeightComputer_40802189312445
// MI455X (gfx1250) — compile-verified
//
#include <hip/hip_runtime.h>
#include <stdint.h>

// Problem constants (reference: y_d, y_gt are (32,1,768,768) f32)
#define N_IMG   32
#define HH      768
#define WW      768
#define HW      (HH * WW)          // 589824
#define TX      64                 // tile width  (output)
#define TY      32                 // tile height (output)
#define HALO    2
#define LW      (TX + 2 * HALO)    // 68  LDS halo width
#define LH      (TY + 2 * HALO)    // 36  LDS halo height
#define TILES_X (WW / TX)          // 12
#define TILES_Y (HH / TY)          // 24
#define NPART   (TILES_X * TILES_Y)// 288 partial sums per image
#define INV_LOG2 1.4426950408889634f

typedef __attribute__((address_space(1))) int g_i32;
typedef __attribute__((address_space(3))) int l_i32;

// ---------------------------------------------------------------------------
// CDNA5 async global->LDS copy (ASYNCcnt path). Builtin if the toolchain has
// it, otherwise raw gfx1250 mnemonics via inline asm (ISA 15.18.3 op 96).
// ---------------------------------------------------------------------------
__device__ __forceinline__ void async_copy_f32(const float* gsrc, float* ldst) {
#if __has_builtin(__builtin_amdgcn_global_load_async_to_lds_b32)
    __builtin_amdgcn_global_load_async_to_lds_b32(
        (g_i32*)(void*)gsrc,
        (l_i32*)(void*)ldst,
        0, 0);
#else
    unsigned lds_off = (unsigned)(uintptr_t)(l_i32*)(void*)ldst;
    unsigned long long ga = (unsigned long long)(uintptr_t)gsrc;
    asm volatile("global_load_async_to_lds_b32 %0, %1, off"
                 :: "v"(lds_off), "v"(ga) : "memory");
#endif
}

__device__ __forceinline__ void wait_async0() {
#if __has_builtin(__builtin_amdgcn_s_wait_asynccnt)
    __builtin_amdgcn_s_wait_asynccnt(0);
#else
    asm volatile("s_wait_asynccnt 0" ::: "memory");
#endif
}

// ---------------------------------------------------------------------------
// Pass 1: fused sigmoid + separable 5x5 stencil + entropy + weight.
// Writes un-normalized w to `out`, one deterministic partial sum per block.
// grid = (12, 24, 32), block = 256 threads (8 waves).
// ---------------------------------------------------------------------------
__global__ __launch_bounds__(256)
void wc_main(const float* __restrict__ yd, const float* __restrict__ ygt,
             float* __restrict__ out, float* __restrict__ partials) {
    __shared__ float raw_[LH * LW];   // logits (halo tile)
    __shared__ float sp_ [LH * LW];   // p  = sigmoid(y), 0 outside image
    __shared__ float sq_ [LH * LW];   // p^2
    __shared__ float h1_ [LH * TX];   // horizontal 5-sum of p
    __shared__ float h2_ [LH * TX];   // horizontal 5-sum of p^2
    __shared__ float red_[256];

    const int tid = threadIdx.x;
    const int x0  = blockIdx.x * TX;
    const int y0  = blockIdx.y * TY;
    const int img = blockIdx.z;
    const long imgbase = (long)img * HW;

    const int xo = tid & (TX - 1);   // output-x owned by this thread
    const int yg = tid >> 6;         // 0..3 -> rows yg*8 .. yg*8+7

    // Prefetch the y_gt lines this thread will consume in phase D.
    #pragma unroll
    for (int r = 0; r < 8; ++r) {
        int gy = y0 + yg * 8 + r;
        __builtin_prefetch(&ygt[imgbase + (long)gy * WW + x0 + xo], 0, 3);
    }

    // Phase A: async-stage the raw halo tile (in-bounds cells only).
    for (int i = tid; i < LH * LW; i += 256) {
        int ly = i / LW, lx = i - ly * LW;
        int gy = y0 - HALO + ly, gx = x0 - HALO + lx;
        if (gy >= 0 && gy < HH && gx >= 0 && gx < WW)
            async_copy_f32(&yd[imgbase + (long)gy * WW + gx], &raw_[i]);
    }
    wait_async0();   // this wave's own copies done; same i-mapping used below

    // Phase B: sigmoid (zero-padded outside the image) + squares.
    for (int i = tid; i < LH * LW; i += 256) {
        int ly = i / LW, lx = i - ly * LW;
        int gy = y0 - HALO + ly, gx = x0 - HALO + lx;
        float p = 0.f;
        if (gy >= 0 && gy < HH && gx >= 0 && gx < WW)
            p = 1.0f / (1.0f + __expf(-raw_[i]));
        sp_[i] = p;
        sq_[i] = p * p;
    }
    __syncthreads();

    // Phase C: horizontal 5-wide box sums for all halo rows.
    for (int i = tid; i < LH * TX; i += 256) {
        int ly = i >> 6, lx = i & (TX - 1);
        int b = ly * LW + lx;
        h1_[i] = sp_[b] + sp_[b+1] + sp_[b+2] + sp_[b+3] + sp_[b+4];
        h2_[i] = sq_[b] + sq_[b+1] + sq_[b+2] + sq_[b+3] + sq_[b+4];
    }
    __syncthreads();

    // Phase D: sliding vertical 5-sum; fuse entropy + weight; store.
    const int ys = yg * 8;
    float s1 = 0.f, s2 = 0.f;
    #pragma unroll
    for (int k = 0; k < 5; ++k) {
        s1 += h1_[(ys + k) * TX + xo];
        s2 += h2_[(ys + k) * TX + xo];
    }
    float lsum = 0.f;
    #pragma unroll
    for (int r = 0; r < 8; ++r) {
        const int y  = ys + r;
        const int c  = (y + HALO) * LW + (xo + HALO);
        const float p  = sp_[c];
        const float p2 = sq_[c];
        const float yv = raw_[c];
        const float S1 = s1 - p;            // sum of valid-neighbor p
        const float S2 = s2 - p2;           // sum of valid-neighbor p^2
        const int gy = y0 + y, gx = x0 + xo;
        const int vy = min(gy + 2, HH - 1) - max(gy - 2, 0) + 1;
        const int vx = min(gx + 2, WW - 1) - max(gx - 2, 0) + 1;
        const float dv  = (float)(vy * vx - 1);
        const float acc = dv * p2 - 2.0f * p * S1 + S2;
        const float cons = 1.0f - acc / dv;
        const float spv = fmaxf(yv, 0.f) + __logf(1.0f + __expf(-fabsf(yv)));
        const float ent = 1.0f + (p * yv - spv) * INV_LOG2;
        const long gidx = imgbase + (long)gy * WW + gx;
        float w = fmaxf(cons * ent, ygt[gidx]);
        w = 0.9f * w + 0.1f;
        out[gidx] = w;
        lsum += w;
        if (r < 7) {
            s1 += h1_[(y + 5) * TX + xo] - h1_[y * TX + xo];
            s2 += h2_[(y + 5) * TX + xo] - h2_[y * TX + xo];
        }
    }

    // Deterministic block reduction -> one partial per tile.
    red_[tid] = lsum;
    __syncthreads();
    for (int s = 128; s > 0; s >>= 1) {
        if (tid < s) red_[tid] += red_[tid + s];
        __syncthreads();
    }
    if (tid == 0)
        partials[(img * TILES_Y + blockIdx.y) * TILES_X + blockIdx.x] = red_[0];
}

// ---------------------------------------------------------------------------
// Pass 2: reduce 288 partials/image -> scale = HW / sum. grid = 32 blocks.
// ---------------------------------------------------------------------------
__global__ __launch_bounds__(256)
void wc_reduce(const float* __restrict__ partials, float* __restrict__ scales) {
    __shared__ float red[256];
    const int img = blockIdx.x, t = threadIdx.x;
    const float* p = partials + img * NPART;
    float s = p[t];
    if (t < NPART - 256) s += p[256 + t];
    red[t] = s;
    __syncthreads();
    for (int st = 128; st > 0; st >>= 1) {
        if (t < st) red[t] += red[t + st];
        __syncthreads();
    }
    if (t == 0) scales[img] = (float)HW / red[0];
}

// ---------------------------------------------------------------------------
// Pass 3: w *= HW / sum  (float4 vectorized; intermediate lives in L2).
// ---------------------------------------------------------------------------
__global__ __launch_bounds__(256)
void wc_normalize(float4* __restrict__ out, const float* __restrict__ scales,
                  int n4) {
    const int stride = gridDim.x * blockDim.x;
    for (int i = blockIdx.x * blockDim.x + threadIdx.x; i < n4; i += stride) {
        const int img = i / (HW / 4);
        const float s = scales[img];
        float4 v = out[i];
        v.x *= s; v.y *= s; v.z *= s; v.w *= s;
        out[i] = v;
    }
}

// ---------------------------------------------------------------------------
extern "C" void kernel_launch(void* const* d_in, const int* in_sizes, int n_in,
                              void* d_out, int out_size, void* d_ws, size_t ws_size,
                              hipStream_t stream) {
    (void)in_sizes; (void)n_in; (void)out_size; (void)ws_size;
    const float* y_d  = (const float*)d_in[0];
    const float* y_gt = (const float*)d_in[1];
    float* out = (float*)d_out;

    float* partials = (float*)d_ws;                 // 32*288 floats
    float* scales   = partials + N_IMG * NPART;     // 32 floats

    dim3 grid1(TILES_X, TILES_Y, N_IMG);
    wc_main<<<grid1, 256, 0, stream>>>(y_d, y_gt, out, partials);
    wc_reduce<<<N_IMG, 256, 0, stream>>>(partials, scales);
    const int n4 = N_IMG * HW / 4;
    wc_normalize<<<2304, 256, 0, stream>>>((float4*)out, scales, n4);
}